// RTMDet_35862976922190
// MI455X (gfx1250) — compile-verified
//
#include <hip/hip_runtime.h>
#include <stdint.h>

typedef unsigned int v4u __attribute__((ext_vector_type(4)));
typedef int          v8i __attribute__((ext_vector_type(8)));
typedef int          v4i __attribute__((ext_vector_type(4)));

#define TILE  128          // anchors per workgroup tile
#define NCLS  80           // classes
#define NB    64           // batch
#define NTOT  8400         // anchors per image (6400 + 1600 + 400)
#define TILES_PER_B 67     // 50 + 13 + 4

__global__ __launch_bounds__(TILE) void rtmdet_decode_kernel(
    const float* __restrict__ cls0, const float* __restrict__ box0,
    const float* __restrict__ cls1, const float* __restrict__ box1,
    const float* __restrict__ cls2, const float* __restrict__ box2,
    float* __restrict__ out)
{
    // [NCLS rows][tileW cols] staged by the Tensor Data Mover
    __shared__ float smem[NCLS * TILE];   // 40 KB

    const int bid = blockIdx.x;
    const int b   = bid / TILES_PER_B;
    const int r   = bid - b * TILES_PER_B;

    const float* cls; const float* box;
    int W, HW, step, off, tileIdx;
    if (r < 50)      { cls = cls0; box = box0; W = 80; HW = 6400; step = 8;  off = 0;    tileIdx = r;      }
    else if (r < 63) { cls = cls1; box = box1; W = 40; HW = 1600; step = 16; off = 6400; tileIdx = r - 50; }
    else             { cls = cls2; box = box2; W = 20; HW = 400;  step = 32; off = 8000; tileIdx = r - 63; }

    const int hw0   = tileIdx * TILE;
    int tileW = HW - hw0; if (tileW > TILE) tileW = TILE;

    const int t = threadIdx.x;
    const float* bp = box + (size_t)b * 4 * HW + (hw0 + t);   // per-thread box column

    // ---- TDM: async DMA of the strided [80 x tileW] cls tile into LDS ----
    // D# per CDNA5 ISA ch.7/8: 2D tensor, data_size=4B, dim0 stride = HW.
    if (t < 32) {   // one wave issues the descriptor
        const float* gptr = cls + (size_t)b * NCLS * HW + hw0;   // tile start
        const uint64_t ga = (uint64_t)(uintptr_t)gptr;
        const uint32_t lds = (uint32_t)(uintptr_t)(void*)smem;   // low 32 bits = LDS byte offset

        v4u g0;
        g0.x = 1u;                                   // count=1 (valid), user mode
        g0.y = lds;                                  // lds_addr
        g0.z = (uint32_t)ga;                         // global_addr[31:0]
        g0.w = (uint32_t)((ga >> 32) & 0x01FFFFFFull) | (2u << 30); // ga[56:32] | type=2

        const uint32_t td0 = (uint32_t)tileW;        // tensor_dim0 (all accesses in-bounds)
        const uint32_t td1 = (uint32_t)NCLS;         // tensor_dim1 = 80 rows
        v8i g1;
        g1[0] = (int)(2u << 16);                                   // data_size=4B; mask=0; no pad/iter
        g1[1] = (int)((td0 & 0xFFFFu) << 16);                      // tensor_dim0[15:0] @ bits 63:48
        g1[2] = (int)((td0 >> 16) | ((td1 & 0xFFFFu) << 16));      // td0[31:16] | td1[15:0]
        g1[3] = (int)((td1 >> 16) | ((uint32_t)tileW << 16));      // td1[31:16] | tile_dim0
        g1[4] = (int)NCLS;                                         // tile_dim1=80, tile_dim2=0
        g1[5] = (int)HW;                                           // tensor_dim0_stride[31:0]
        g1[6] = 0;                                                 // stride0[47:32]=0, stride1[15:0]=0
        g1[7] = 0;

        v4i gz = {0, 0, 0, 0};
#if defined(__clang_major__) && (__clang_major__ >= 23)
        v8i gz8 = {0, 0, 0, 0, 0, 0, 0, 0};
        __builtin_amdgcn_tensor_load_to_lds(g0, g1, gz, gz, gz8, 0);
#else
        __builtin_amdgcn_tensor_load_to_lds(g0, g1, gz, gz, 0);
#endif
        __builtin_amdgcn_s_wait_tensorcnt(0);        // DMA complete before barrier
    } else if (t < tileW) {
        // Overlap: warm the 4 strided box planes while the TDM streams cls.
        __builtin_prefetch(bp,          0, 3);       // -> global_prefetch_b8
        __builtin_prefetch(bp + HW,     0, 3);
        __builtin_prefetch(bp + 2 * HW, 0, 3);
        __builtin_prefetch(bp + 3 * HW, 0, 3);
    }
    __syncthreads();

    if (t < tileW) {
        // ---- max/argmax over 80 classes from LDS (conflict-free columns) ----
        const float* col = smem + t;
        float m = col[0];
        int   mi = 0;
        #pragma unroll 8
        for (int c = 1; c < NCLS; ++c) {
            float v = col[c * tileW];
            if (v > m) { m = v; mi = c; }   // strict '>' keeps first index (argmax semantics)
        }
        const float conf = 1.0f / (1.0f + __expf(-m));   // sigmoid of the max logit

        // ---- box decode ----
        const int a = hw0 + t;
        const float d0 = bp[0];
        const float d1 = bp[HW];
        const float d2 = bp[2 * HW];
        const float d3 = bp[3 * HW];

        const int   x  = a % W;
        const int   y  = a / W;
        const float gx = (float)(x * step);
        const float gy = (float)(y * step);

        const float hi = 639.0f;
        float x1 = fminf(fmaxf(gx - d0, 0.0f), hi);
        float y1 = fminf(fmaxf(gy - d1, 0.0f), hi);
        float x2 = fminf(fmaxf(gx + d2, 0.0f), hi);
        float y2 = fminf(fmaxf(gy + d3, 0.0f), hi);

        const size_t n = (size_t)b * NTOT + (size_t)off + (size_t)a;
        float4* ob = (float4*)out;                       // boxes: [NB*NTOT] float4
        ob[n] = make_float4(x1, y1, x2, y2);             // single b128 store

        float* os = out + (size_t)NB * NTOT * 4;         // scores
        os[n] = conf;
        float* oc = os + (size_t)NB * NTOT;              // classes (as float)
        oc[n] = (float)mi;
    }
}

extern "C" void kernel_launch(void* const* d_in, const int* in_sizes, int n_in,
                              void* d_out, int out_size, void* d_ws, size_t ws_size,
                              hipStream_t stream) {
    const float* cls0 = (const float*)d_in[0];
    const float* box0 = (const float*)d_in[1];
    const float* cls1 = (const float*)d_in[2];
    const float* box1 = (const float*)d_in[3];
    const float* cls2 = (const float*)d_in[4];
    const float* box2 = (const float*)d_in[5];
    float* out = (float*)d_out;

    dim3 grid(NB * TILES_PER_B);   // 4288 blocks
    dim3 block(TILE);              // 128 threads = 4 waves
    rtmdet_decode_kernel<<<grid, block, 0, stream>>>(cls0, box0, cls1, box1, cls2, box2, out);
}